// DCLLoss_1614907703520
// MI455X (gfx1250) — compile-verified
//
#include <hip/hip_runtime.h>
#include <math.h>

// ---------------------------------------------------------------------------
// DCL loss, fused GEMM + online logsumexp on CDNA5 (gfx1250).
//  * f32 scores via V_WMMA_F32_16X16X4_F32 (full fp32 accuracy, matrix pipe)
//  * 2 column tiles per wave iteration -> A-fragment reused by 2 WMMAs
//  * B staging via GLOBAL_LOAD_ASYNC_TO_LDS_B128, double-buffered K chunks,
//    synchronized with s_wait_asynccnt (ASYNCcnt completes in order)
//  * never materializes the [N,2N] similarity matrix (it would be 512 MB)
// ---------------------------------------------------------------------------

#define NROWS   8192
#define DDIM    1024
#define INV_T   10.0f
#define SMALLN  (-103.61632918473207f)   // log(1e-45)
#define KCC     64                        // K chunk per staged buffer

typedef __attribute__((ext_vector_type(2))) float v2f;
typedef __attribute__((ext_vector_type(8))) float v8f;

#define A_STRIDE 1028                     // 1024+4: 4*m mod 64 -> distinct banks
#define B_STRIDE 68                       // 64+4:   4*n mod 64 -> distinct banks
#define A_FLOATS (16 * A_STRIDE)          // 16448
#define B_BUF_FLOATS (32 * B_STRIDE)      // one chunk, 32 cols (2 tiles): 2176
#define B_WAVE_FLOATS (2 * B_BUF_FLOATS)  // double buffered: 4352
#define SMEM_FLOATS (A_FLOATS + 4 * B_WAVE_FLOATS + 64 + 64 + 16)  // 34000

__device__ __forceinline__ unsigned lds_off(const void* p) {
  // generic (flat) LDS pointer: addr[31:0] is the wave-relative LDS address
  return (unsigned)(size_t)p;
}

__global__ __launch_bounds__(128) void dcl_fused(const float* __restrict__ z1,
                                                 const float* __restrict__ z2,
                                                 float* __restrict__ partial) {
  extern __shared__ float smem[];
  float* As   = smem;                               // [16][A_STRIDE]
  float* Ball = smem + A_FLOATS;                    // 4 wave-private dbl buffers
  float* redM = Ball + 4 * B_WAVE_FLOATS;           // [4*16]
  float* redS = redM + 64;                          // [4*16]
  float* pos  = redS + 64;                          // [16] positive terms

  const int tid     = threadIdx.x;
  const int wave    = tid >> 5;
  const int lane    = tid & 31;
  const int rowBase = blockIdx.x * 16;

  // ---- cooperative A tile load: 16 rows x 1024 f32, padded rows ----
  for (int i = tid; i < (16 * DDIM) / 4; i += 128) {
    int r = i >> 8;            // 256 float4 per row
    int c = i & 255;
    float4 v = ((const float4*)(z1 + (size_t)(rowBase + r) * DDIM))[c];
    ((float4*)(As + r * A_STRIDE))[c] = v;
  }
  __syncthreads();

  float* Bl = Ball + wave * B_WAVE_FLOATS;
  const unsigned BlOff = lds_off(Bl);

  // A-fragment (16x4 f32): lanes 0-15 M=0..15 K={k,k+1}; lanes 16-31 K={k+2,k+3}
  const int mA   = lane & 15;
  const int koff = (lane >> 4) * 2;
  const float* aRow = As + mA * A_STRIDE + koff;
  // C/D layout: lane l holds col n=l&15, rows m = r + 8*(l>>4)
  const int nC  = lane & 15;
  const int mhi = (lane >> 4) * 8;

  // async staging: lane covers col (2q + lane>>4), float4 slot (lane&15)
  const int stgCol  = lane >> 4;          // 0 or 1
  const int stgSlot = lane & 15;          // float4 index within the chunk
  const unsigned ldsLaneOff = BlOff + (unsigned)stgCol * (B_STRIDE * 4)
                                    + (unsigned)stgSlot * 16;

  float mx[8], sm[8];
#pragma unroll
  for (int r = 0; r < 8; ++r) { mx[r] = -1.0e30f; sm[r] = 0.0f; }

  // ---- stream 512 column tile-pairs (32 cols each), 4 waves round-robin ----
  for (int p = wave; p < 512; p += 4) {
    const int colBase = p << 5;                      // 32-col aligned
    const float* mat;
    int cb;
    if (colBase < NROWS) { mat = z1; cb = colBase; }           // self-sim half
    else                 { mat = z2; cb = colBase - NROWS; }   // cross half

    // per-lane global base for staging (element units)
    const float* gLane = mat + (size_t)(cb + stgCol) * DDIM + (size_t)stgSlot * 4;

    // prologue: stage chunk 0 into buffer 0 (16 async b128 per lane)
#pragma unroll
    for (int q = 0; q < 16; ++q) {
      asm volatile("global_load_async_to_lds_b128 %0, %1, off"
                   :: "v"(ldsLaneOff + (unsigned)q * (2 * B_STRIDE * 4)),
                      "v"(gLane + (size_t)(2 * q) * DDIM)
                   : "memory");
    }

    v8f acc0 = {}, acc1 = {};
    for (int c = 0; c < DDIM / KCC; ++c) {
      if (c + 1 < DDIM / KCC) {
        // stage next chunk into the other buffer
        const unsigned nb = ((unsigned)(c + 1) & 1u) * (B_BUF_FLOATS * 4);
        const float* gNext = gLane + (size_t)(c + 1) * KCC;
#pragma unroll
        for (int q = 0; q < 16; ++q) {
          asm volatile("global_load_async_to_lds_b128 %0, %1, off"
                       :: "v"(ldsLaneOff + nb + (unsigned)q * (2 * B_STRIDE * 4)),
                          "v"(gNext + (size_t)(2 * q) * DDIM)
                       : "memory");
        }
        asm volatile("s_wait_asynccnt 0x10" ::: "memory");  // drain chunk c
      } else {
        asm volatile("s_wait_asynccnt 0x0" ::: "memory");
      }

      const float* buf = Bl + (c & 1) * B_BUF_FLOATS;
      const float* bR0 = buf + nC * B_STRIDE + koff;
      const float* bR1 = buf + (16 + nC) * B_STRIDE + koff;
      const float* aK  = aRow + c * KCC;
#pragma unroll
      for (int k = 0; k < KCC; k += 4) {
        v2f a  = *(const v2f*)(aK + k);
        v2f b0 = *(const v2f*)(bR0 + k);
        v2f b1 = *(const v2f*)(bR1 + k);
        acc0 = __builtin_amdgcn_wmma_f32_16x16x4_f32(
            false, a, false, b0, (short)0, acc0, false, false);
        acc1 = __builtin_amdgcn_wmma_f32_16x16x4_f32(
            false, a, false, b1, (short)0, acc1, false, false);
      }
    }

    // ---- scale, mask diagonals, online logsumexp (both tiles) ----
#pragma unroll
    for (int tt = 0; tt < 2; ++tt) {
      const int j = colBase + tt * 16 + nC;
#pragma unroll
      for (int r = 0; r < 8; ++r) {
        const int grow = rowBase + mhi + r;
        float x = (tt ? acc1[r] : acc0[r]) * INV_T;
        if (j == grow + NROWS) pos[mhi + r] = -x;   // positive term (pre-mask)
        if (j == grow || j == grow + NROWS) x += SMALLN;
        if (x > mx[r]) {
          sm[r] = sm[r] * __expf(mx[r] - x) + 1.0f;
          mx[r] = x;
        } else {
          sm[r] += __expf(x - mx[r]);
        }
      }
    }
  }

  // ---- merge across the 16 column-lanes (butterfly within each half) ----
#pragma unroll
  for (int off = 1; off < 16; off <<= 1) {
#pragma unroll
    for (int r = 0; r < 8; ++r) {
      float om = __shfl_xor(mx[r], off, 32);
      float os = __shfl_xor(sm[r], off, 32);
      float M  = fmaxf(mx[r], om);
      sm[r] = sm[r] * __expf(mx[r] - M) + os * __expf(om - M);
      mx[r] = M;
    }
  }
  if ((lane & 15) == 0) {
#pragma unroll
    for (int r = 0; r < 8; ++r) {
      redM[wave * 16 + mhi + r] = mx[r];
      redS[wave * 16 + mhi + r] = sm[r];
    }
  }
  __syncthreads();

  // ---- merge across the 4 waves, finish rows, sum stripe, emit partial ----
  if (tid < 16) {
    float M = redM[tid], S = redS[tid];
#pragma unroll
    for (int w = 1; w < 4; ++w) {
      float om = redM[w * 16 + tid], os = redS[w * 16 + tid];
      float Mn = fmaxf(M, om);
      S = S * __expf(M - Mn) + os * __expf(om - Mn);
      M = Mn;
    }
    float loss = pos[tid] + M + logf(S);
    loss += __shfl_xor(loss, 1, 32);
    loss += __shfl_xor(loss, 2, 32);
    loss += __shfl_xor(loss, 4, 32);
    loss += __shfl_xor(loss, 8, 32);
    if (tid == 0) partial[blockIdx.x] = loss * (1.0f / (float)NROWS);
  }
}

// Deterministic fixed-order tree reduction of the 512 per-stripe partials.
__global__ __launch_bounds__(512) void dcl_reduce(const float* __restrict__ partial,
                                                  float* __restrict__ out) {
  __shared__ float buf[512];
  const int tid = threadIdx.x;
  buf[tid] = partial[tid];
  __syncthreads();
  for (int s = 256; s > 0; s >>= 1) {
    if (tid < s) buf[tid] += buf[tid + s];
    __syncthreads();
  }
  if (tid == 0) out[0] = buf[0];
}

extern "C" void kernel_launch(void* const* d_in, const int* in_sizes, int n_in,
                              void* d_out, int out_size, void* d_ws, size_t ws_size,
                              hipStream_t stream) {
  const float* z1 = (const float*)d_in[0];
  const float* z2 = (const float*)d_in[1];
  float* out      = (float*)d_out;
  float* partial  = (float*)d_ws;   // 512 floats of scratch

  const size_t smemBytes = (size_t)SMEM_FLOATS * sizeof(float);  // 136 KB dynamic LDS
  hipFuncSetAttribute(reinterpret_cast<const void*>(dcl_fused),
                      hipFuncAttributeMaxDynamicSharedMemorySize, (int)smemBytes);

  dcl_fused<<<512, 128, smemBytes, stream>>>(z1, z2, partial);
  dcl_reduce<<<1, 512, 0, stream>>>(partial, out);
}